// GCNConv_8366596292669
// MI455X (gfx1250) — compile-verified
//
#include <hip/hip_runtime.h>

typedef __attribute__((ext_vector_type(2))) float v2f;
typedef __attribute__((ext_vector_type(8))) float v8f;

#define C_DIM 128

// ---------------------------------------------------------------------------
// Kernel 1: h[row, :] = (x[row, :] @ W) * norm[row]
// 256 threads = 8 waves; wave w computes rows [blockRow + 16w, +16).
// W staged in LDS in a k-pair swizzled layout (64 KB):
//   dword(p, col) = p*256 + ((2*col + 32*(p&1)) & 255), p = k>>1
// so each B fragment (values k, k+1 for one column) is ONE aligned
// ds_load_b64, and the two lane-halves (p vs p+1) hit disjoint bank sets.
// ---------------------------------------------------------------------------
__global__ __launch_bounds__(256) void gcn_gemm_norm(
    const float* __restrict__ x, const float* __restrict__ w,
    const float* __restrict__ norm, float* __restrict__ h, int N)
{
    __shared__ float lds_w[C_DIM * C_DIM];  // 64 KB, swizzled k-pair layout

    // Cooperative staging: idx -> (p = idx>>7, n = idx&127); gather rows 2p,2p+1.
    for (int idx = threadIdx.x; idx < (C_DIM / 2) * C_DIM; idx += 256) {
        const int p = idx >> 7;
        const int n = idx & 127;
        v2f v;
        v.x = w[(2 * p + 0) * C_DIM + n];
        v.y = w[(2 * p + 1) * C_DIM + n];
        const int dw = p * 256 + ((2 * n + 32 * (p & 1)) & 255);
        *(v2f*)&lds_w[dw] = v;
    }
    __syncthreads();

    const int wave   = threadIdx.x >> 5;
    const int lane   = threadIdx.x & 31;
    const int lane16 = lane & 15;
    const int khalf  = lane >> 4;          // 0 for lanes 0-15, 1 for 16-31

    const int m0 = (blockIdx.x * 8 + wave) * 16;    // base output row of this wave
    int row  = m0 + lane16;
    int rowc = row < N ? row : N - 1;               // clamp: keep EXEC all-ones

    v8f acc[8];
    const v8f vzero = {0.f,0.f,0.f,0.f,0.f,0.f,0.f,0.f};
    #pragma unroll
    for (int n = 0; n < 8; ++n) acc[n] = vzero;

    const float* xrow = x + (size_t)rowc * C_DIM;

    // Per-n LDS offsets are k-invariant (p&1 == khalf for k % 4 == 0).
    int coff[8];
    #pragma unroll
    for (int n = 0; n < 8; ++n)
        coff[n] = (32 * n + 2 * lane16 + 32 * khalf) & 255;
    const float* lwp = lds_w + khalf * 256;  // + (k>>1)*256 per step

    // K loop: 128 = 32 steps of K=4.
    // A frag (f32 16x4): lane = row (lane&15); VGPR0/1 = K = kb, kb+1 (kb = k+2*khalf)
    // B frag (f32 4x16): lane = col (lane&15); VGPR0/1 = K = kb, kb+1
    #pragma unroll 4
    for (int k = 0; k < C_DIM; k += 4) {
        const int kb = k + 2 * khalf;
        v2f a = *(const v2f*)(xrow + kb);
        const float* lk = lwp + (k >> 1) * 256;
        #pragma unroll
        for (int n = 0; n < 8; ++n) {
            v2f b = *(const v2f*)(lk + coff[n]);
            acc[n] = __builtin_amdgcn_wmma_f32_16x16x4_f32(
                false, a, false, b, (short)0, acc[n], false, false);
        }
    }

    // C/D layout: vector element j <-> output row m0 + j + 8*khalf; col = n*16 + lane16
    float nrm[8];
    #pragma unroll
    for (int j = 0; j < 8; ++j) {
        const int r = m0 + j + 8 * khalf;
        nrm[j] = (r < N) ? norm[r] : 0.0f;
    }
    #pragma unroll
    for (int n = 0; n < 8; ++n) {
        #pragma unroll
        for (int j = 0; j < 8; ++j) {
            const int r = m0 + j + 8 * khalf;
            if (r < N) {
                h[(size_t)r * C_DIM + n * 16 + lane16] = acc[n][j] * nrm[j];
            }
        }
    }
}

// ---------------------------------------------------------------------------
// Kernel 2: zero the output accumulator (d_out is poisoned by the harness)
// ---------------------------------------------------------------------------
__global__ __launch_bounds__(256) void gcn_zero(float4* __restrict__ out, int n4)
{
    int i = blockIdx.x * blockDim.x + threadIdx.x;
    if (i < n4) out[i] = make_float4(0.f, 0.f, 0.f, 0.f);
}

// ---------------------------------------------------------------------------
// Kernel 3: one wave per edge; lane l handles channels [4l, 4l+4).
// Gather h[src] (512B coalesced, L2-resident) -> 4 f32 atomics into out[dst].
// ---------------------------------------------------------------------------
__global__ __launch_bounds__(256) void gcn_scatter(
    const float* __restrict__ h, const int* __restrict__ esrc,
    const int* __restrict__ edst, float* __restrict__ out, int E)
{
    const long long tid = (long long)blockIdx.x * blockDim.x + threadIdx.x;
    const long long e = tid >> 5;
    if (e >= E) return;
    const int lane = (int)(tid & 31);
    const int src = esrc[e];
    const int dst = edst[e];
    const float4 v = *(const float4*)(h + (size_t)src * C_DIM + lane * 4);
    float* o = out + (size_t)dst * C_DIM + lane * 4;
    atomicAdd(o + 0, v.x);
    atomicAdd(o + 1, v.y);
    atomicAdd(o + 2, v.z);
    atomicAdd(o + 3, v.w);
}

// ---------------------------------------------------------------------------
// Kernel 4: out = out * norm[row] + bias[col], in place, float4 granularity
// ---------------------------------------------------------------------------
__global__ __launch_bounds__(256) void gcn_final(
    float* __restrict__ out, const float* __restrict__ norm,
    const float* __restrict__ bias, int N)
{
    const long long tid = (long long)blockIdx.x * blockDim.x + threadIdx.x;
    if (tid >= (long long)N * 32) return;
    const int row = (int)(tid >> 5);
    const int c4  = (int)(tid & 31) * 4;
    const float nm = norm[row];
    float4 v = ((float4*)out)[tid];
    const float4 b = *(const float4*)(bias + c4);
    v.x = v.x * nm + b.x;
    v.y = v.y * nm + b.y;
    v.z = v.z * nm + b.z;
    v.w = v.w * nm + b.w;
    ((float4*)out)[tid] = v;
}

extern "C" void kernel_launch(void* const* d_in, const int* in_sizes, int n_in,
                              void* d_out, int out_size, void* d_ws, size_t ws_size,
                              hipStream_t stream) {
    const float* x    = (const float*)d_in[0];
    const float* norm = (const float*)d_in[1];
    const float* w    = (const float*)d_in[2];
    const float* bias = (const float*)d_in[3];
    const int*   esrc = (const int*)d_in[4];
    const int*   edst = (const int*)d_in[5];
    float* out = (float*)d_out;

    const int N = in_sizes[0] / C_DIM;
    const int E = in_sizes[4];
    float* h = (float*)d_ws;   // N*128 f32 = 51.2 MB scratch

    dim3 blk(256);

    // 1) h = (x @ W) * norm   (WMMA f32)
    gcn_gemm_norm<<<(N + 127) / 128, blk, 0, stream>>>(x, w, norm, h, N);

    // 2) zero output accumulator
    const int n4 = out_size / 4;
    gcn_zero<<<(n4 + 255) / 256, blk, 0, stream>>>((float4*)out, n4);

    // 3) edge gather + scatter-add (one wave per edge)
    const long long sthreads = (long long)E * 32;
    gcn_scatter<<<(unsigned)((sthreads + 255) / 256), blk, 0, stream>>>(h, esrc, edst, out, E);

    // 4) out = out*norm + bias
    const long long fthreads = (long long)N * 32;
    gcn_final<<<(unsigned)((fthreads + 255) / 256), blk, 0, stream>>>(out, norm, bias, N);
}